// OptMapsSheafLearner_8976481648850
// MI455X (gfx1250) — compile-verified
//
#include <hip/hip_runtime.h>
#include <hip/hip_bf16.h>
#include <math.h>

// ---------------------------------------------------------------------------
// GGW mapping kernel for MI455X (gfx1250, wave32).
//   per edge: sig_r=x_sig[row], sig_c=x_sig[col]; eigh both; A = Ps diag(sqrt(Lt/Ls)) Pt^T
// Final 4x4 products batched 4-edges-per-WMMA via v_wmma_f32_16x16x4_f32,
// computed TRANSPOSED (Pt*diag(s) x Ps^T) so each lane owns a contiguous output
// row -> single b128 nontemporal store per group, 256B-contiguous across lanes.
// ---------------------------------------------------------------------------

typedef float v2f __attribute__((ext_vector_type(2)));
typedef float v4f __attribute__((ext_vector_type(4)));
typedef float v8f __attribute__((ext_vector_type(8)));

#define LDS_STRIDE 20  // 16 payload floats + 4 pad (16B-aligned b128 writes, de-conflicted banks)

// Fully-unrolled cyclic Jacobi eigensolver for symmetric 4x4.
// Outputs ascending eigenvalues lv[4] and eigenvectors column-major: vc[j][i] = V[i][j].
__device__ __forceinline__ void eigh4(const float in[16], float lv[4], float vc[4][4]) {
  float a[4][4];
#pragma unroll
  for (int i = 0; i < 4; ++i)
#pragma unroll
    for (int j = 0; j < 4; ++j) a[i][j] = in[i * 4 + j];
#pragma unroll
  for (int j = 0; j < 4; ++j)
#pragma unroll
    for (int i = 0; i < 4; ++i) vc[j][i] = (i == j) ? 1.0f : 0.0f;

  const int PP[6] = {0, 0, 0, 1, 1, 2};
  const int QQ[6] = {1, 2, 3, 2, 3, 3};
#pragma unroll
  for (int sweep = 0; sweep < 5; ++sweep) {
#pragma unroll
    for (int r = 0; r < 6; ++r) {
      const int p = PP[r], q = QQ[r];
      const float apq = a[p][q];
      float c = 1.0f, s = 0.0f;
      if (fabsf(apq) > 1e-30f) {
        const float theta = 0.5f * (a[q][q] - a[p][p]) / apq;
        const float t = copysignf(1.0f, theta) / (fabsf(theta) + sqrtf(theta * theta + 1.0f));
        c = rsqrtf(t * t + 1.0f);
        s = t * c;
      }
      // A <- G^T A G ; V <- V G with G[p][p]=c, G[q][q]=c, G[p][q]=s, G[q][p]=-s
#pragma unroll
      for (int k = 0; k < 4; ++k) {
        const float akp = a[k][p], akq = a[k][q];
        a[k][p] = c * akp - s * akq;
        a[k][q] = s * akp + c * akq;
      }
#pragma unroll
      for (int k = 0; k < 4; ++k) {
        const float apk = a[p][k], aqk = a[q][k];
        a[p][k] = c * apk - s * aqk;
        a[q][k] = s * apk + c * aqk;
        const float vkp = vc[p][k], vkq = vc[q][k];
        vc[p][k] = c * vkp - s * vkq;
        vc[q][k] = s * vkp + c * vkq;
      }
    }
  }
  lv[0] = a[0][0]; lv[1] = a[1][1]; lv[2] = a[2][2]; lv[3] = a[3][3];

  // Ascending sort (network for 4); whole eigenvector columns ride along (static indices).
#define CSWAP(I, J)                                                       \
  if (lv[I] > lv[J]) {                                                    \
    float t_ = lv[I]; lv[I] = lv[J]; lv[J] = t_;                          \
    for (int z = 0; z < 4; ++z) {                                         \
      float v_ = vc[I][z]; vc[I][z] = vc[J][z]; vc[J][z] = v_;            \
    }                                                                     \
  }
  CSWAP(0, 1) CSWAP(2, 3) CSWAP(0, 2) CSWAP(1, 3) CSWAP(1, 2)
#undef CSWAP
}

__global__ __launch_bounds__(256) void ggw_maps_kernel(const float* __restrict__ x_sig,
                                                       const int* __restrict__ edge_index,
                                                       float* __restrict__ out, int E) {
  // Per-wave staging for the WMMA operands:
  //   sA row j (per edge): A'[j][k] = Pt[j][k] * s_k      (k-adjacent)
  //   sB row i (per edge): B'[k][i] = Ps[i][k] -> staged as Ps row-major (k-adjacent)
  __shared__ float sA[8 * 32 * LDS_STRIDE];
  __shared__ float sB[8 * 32 * LDS_STRIDE];

  const int tid = threadIdx.x;
  const int lane = tid & 31;
  const int wave = tid >> 5;
  const int blockBase = blockIdx.x * 256;
  const int eGlobal = blockBase + tid;
  // Clamp instead of early-exit: WMMA requires EXEC==all-ones, and __syncthreads
  // must be reached by every thread. Stores are predicated at the end.
  const int e = (eGlobal < E) ? eGlobal : (E - 1);

  const int row = edge_index[e];
  const int col = edge_index[E + e];

  float S[16], T[16];
  const v4f* ps4 = (const v4f*)(x_sig + (size_t)row * 16);
  const v4f* pt4 = (const v4f*)(x_sig + (size_t)col * 16);
#pragma unroll
  for (int i = 0; i < 4; ++i) {
    const v4f r4 = ps4[i];
    S[i * 4 + 0] = r4.x; S[i * 4 + 1] = r4.y; S[i * 4 + 2] = r4.z; S[i * 4 + 3] = r4.w;
    const v4f t4 = pt4[i];
    T[i * 4 + 0] = t4.x; T[i * 4 + 1] = t4.y; T[i * 4 + 2] = t4.z; T[i * 4 + 3] = t4.w;
  }

  float ls[4], lt[4], vs[4][4], vt[4][4];
  eigh4(S, ls, vs);  // Ps (columns = eigenvectors of sig_r): vs[k][i] = Ps[i][k]
  eigh4(T, lt, vt);  // Pt: vt[k][j] = Pt[j][k]

  float sc[4];
#pragma unroll
  for (int j = 0; j < 4; ++j) sc[j] = sqrtf(lt[j] / ls[j]);  // SPD => ls > 0

  float* myA = &sA[(wave * 32 + lane) * LDS_STRIDE];
  float* myB = &sB[(wave * 32 + lane) * LDS_STRIDE];
#pragma unroll
  for (int i = 0; i < 4; ++i) {
    // A'[row j=i][k] = Pt[i][k]*s_k = vt[k][i]*sc[k]
    v4f arow = {vt[0][i] * sc[0], vt[1][i] * sc[1], vt[2][i] * sc[2], vt[3][i] * sc[3]};
    *(v4f*)(myA + i * 4) = arow;
    // B' staged k-adjacent: myB[i*4+k] = Ps[i][k] = vs[k][i]
    v4f brow = {vs[0][i], vs[1][i], vs[2][i], vs[3][i]};
    *(v4f*)(myB + i * 4) = brow;
  }

  __syncthreads();

  const float* wA = &sA[wave * 32 * LDS_STRIDE];
  const float* wB = &sB[wave * 32 * LDS_STRIDE];

  // ISA operand layouts for V_WMMA_F32_16X16X4_F32 (wave32):
  //  A (16x4):  lanes 0-15 hold M=lane with K={0,1} in vgpr{0,1}; lanes 16-31 hold K={2,3}
  //  B (4x16):  vgpr r: lanes 0-15 -> K=r, N=lane; lanes 16-31 -> K=r+2, N=lane-16
  //  C/D(16x16):vgpr r: lanes 0-15 -> M=r, N=lane; lanes 16-31 -> M=r+8, N=lane-16
  const int n = lane & 15;           // A row index m == B/D column index n for this lane
  const int k0 = (lane >> 4) * 2;    // K base for this half-wave

  // Extraction geometry (computed once): D(4e+j, 4e+i) = A_e[i][j], so lane n=4e+i
  // owns output row i of edge e in c[rbase..rbase+3].
  const bool hiHalf = ((n >> 2) & 1) != 0;                 // selects c[4..7] vs c[0..3]
  const bool active = (lane < 8) || (lane >= 24);          // lanes holding diagonal blocks
  const int ebSel = n >> 2;                                // which of the 4 edges in group
  const int irow = n & 3;                                  // output row owned by this lane

#pragma unroll
  for (int g = 0; g < 8; ++g) {  // 8 groups of 4 edges per wave
    const int slotA = g * 4 + (n >> 2);
    v2f a;
    a.x = wA[slotA * LDS_STRIDE + (n & 3) * 4 + k0];
    a.y = wA[slotA * LDS_STRIDE + (n & 3) * 4 + k0 + 1];

    const int slotB = g * 4 + (n >> 2);
    v2f b;
    b.x = wB[slotB * LDS_STRIDE + (n & 3) * 4 + k0];       // B[k][n] = Ps[i][k]
    b.y = wB[slotB * LDS_STRIDE + (n & 3) * 4 + k0 + 1];

    v8f c = {};
    // D = blockstack(Pt_e*diag(s), 16x4) x blockstack(Ps_f^T, 4x16); diag blocks = A_e^T.
    c = __builtin_amdgcn_wmma_f32_16x16x4_f32(false, a, false, b, (short)0, c, false, false);

    // Each active lane selects its 4 consecutive accumulators = one output row.
    v4f rowv;
    rowv.x = hiHalf ? c[4] : c[0];
    rowv.y = hiHalf ? c[5] : c[1];
    rowv.z = hiHalf ? c[6] : c[2];
    rowv.w = hiHalf ? c[7] : c[3];

    const int oe = blockBase + wave * 32 + g * 4 + ebSel;  // global edge id
    if (active && oe < E) {
      // 16 active lanes cover a contiguous 256B (4 edges x 64B) region.
      __builtin_nontemporal_store(rowv, (v4f*)(out + (size_t)oe * 16 + irow * 4));
    }
  }
}

extern "C" void kernel_launch(void* const* d_in, const int* in_sizes, int n_in,
                              void* d_out, int out_size, void* d_ws, size_t ws_size,
                              hipStream_t stream) {
  // inputs: d_in[0]=x_mu (unused by reference output), d_in[1]=x_sig [1,N,4,4] f32,
  //         d_in[2]=edge_index [2,E] i32
  const float* x_sig = (const float*)d_in[1];
  const int* edge_index = (const int*)d_in[2];
  float* out = (float*)d_out;
  const int E = in_sizes[2] / 2;
  const int blocks = (E + 255) / 256;
  ggw_maps_kernel<<<blocks, 256, 0, stream>>>(x_sig, edge_index, out, E);
}